// YoloLoss_80530636800727
// MI455X (gfx1250) — compile-verified
//
#include <hip/hip_runtime.h>

typedef float v2f __attribute__((ext_vector_type(2)));
typedef float v8f __attribute__((ext_vector_type(8)));

#define NBLOCKS 1024
#define NTHREADS 256

// Full 32-lane sum using V_WMMA_F32_16X16X4_F32.
// A (16x4): VGPR0 = x (K0 for lanes 0-15, K2 for lanes 16-31), VGPR1 = 0 (K1/K3).
// B (4x16): all ones.  =>  D[m][n] = x[m] + x[m+16] for every column n.
// D VGPR j holds row j (lanes 0-15) and row j+8 (lanes 16-31); summing the 8
// D VGPRs gives each half-wave the sum of its 8 rows; one shfl_xor(16) merges.
__device__ __forceinline__ float wave_sum_wmma(float x) {
  v2f a; a[0] = x;    a[1] = 0.0f;
  v2f b; b[0] = 1.0f; b[1] = 1.0f;
  v8f c = {};
  c = __builtin_amdgcn_wmma_f32_16x16x4_f32(false, a, false, b, (short)0, c,
                                            false, false);
  float t = ((c[0] + c[1]) + (c[2] + c[3])) + ((c[4] + c[5]) + (c[6] + c[7]));
  t += __shfl_xor(t, 16, 32);
  return t;
}

// Block-level reduction of 4 accumulators -> 4 floats written by thread 0.
__device__ __forceinline__ void block_reduce4(float s0, float s1, float s2,
                                              float s3, float* dst) {
  float r0 = wave_sum_wmma(s0);
  float r1 = wave_sum_wmma(s1);
  float r2 = wave_sum_wmma(s2);
  float r3 = wave_sum_wmma(s3);
  __shared__ float sm[NTHREADS / 32][4];
  const int lane = threadIdx.x & 31;
  const int wid  = threadIdx.x >> 5;
  if (lane == 0) { sm[wid][0] = r0; sm[wid][1] = r1; sm[wid][2] = r2; sm[wid][3] = r3; }
  __syncthreads();
  if (threadIdx.x == 0) {
    float a0 = 0.f, a1 = 0.f, a2 = 0.f, a3 = 0.f;
    for (int i = 0; i < NTHREADS / 32; ++i) {
      a0 += sm[i][0]; a1 += sm[i][1]; a2 += sm[i][2]; a3 += sm[i][3];
    }
    dst[0] = a0; dst[1] = a1; dst[2] = a2; dst[3] = a3;
  }
}

__global__ __launch_bounds__(NTHREADS) void yolo_loss_main(
    const float* __restrict__ pred,        // (cells, 30), rows 8B-aligned
    const float* __restrict__ tbox,        // (cells, 4),  rows 16B-aligned
    const float* __restrict__ tcls,        // (cells, 20), rows 16B-aligned
    const unsigned char* __restrict__ mask,// (cells,)
    float* __restrict__ bsums,             // (gridDim.x, 4)
    int ncells) {
  float s_reg = 0.f, s_con = 0.f, s_no = 0.f, s_cls = 0.f;
  const int stride = gridDim.x * blockDim.x;
  for (int c = blockIdx.x * blockDim.x + threadIdx.x; c < ncells; c += stride) {
    const float2* p2 = reinterpret_cast<const float2*>(pred + (size_t)c * 30);
    // speculative prefetch of next cell's pred row (dropped if OOB)
    __builtin_prefetch(pred + (size_t)(c + stride) * 30, 0, 1);
    const bool m = mask[c] != 0;
    if (m) {
      // wide loads: pred row as 15x float2 (b64), tcls as 5x float4 (b128)
      float pv[30];
#pragma unroll
      for (int k = 0; k < 15; ++k) {
        const float2 t = p2[k];
        pv[2 * k] = t.x; pv[2 * k + 1] = t.y;
      }
      float cv[20];
      const float4* tc4 =
          reinterpret_cast<const float4*>(tcls + (size_t)c * 20);
#pragma unroll
      for (int k = 0; k < 5; ++k) {
        const float4 t = tc4[k];
        cv[4 * k] = t.x; cv[4 * k + 1] = t.y;
        cv[4 * k + 2] = t.z; cv[4 * k + 3] = t.w;
      }
      // class loss
      float cl = 0.f;
#pragma unroll
      for (int k = 0; k < 20; ++k) {
        const float d = pv[10 + k] - cv[k];
        cl += d * d;
      }
      s_cls += cl;
      // target box -> xyxy
      const float4 tb =
          *reinterpret_cast<const float4*>(tbox + (size_t)c * 4);
      const float tx = tb.x, ty = tb.y, tw = tb.z, th = tb.w;
      const float tx1 = tx / 14.0f - 0.5f * tw, ty1 = ty / 14.0f - 0.5f * th;
      const float tx2 = tx / 14.0f + 0.5f * tw, ty2 = ty / 14.0f + 0.5f * th;
      const float area_t = (tx2 - tx1) * (ty2 - ty1);
      float iou[2];
#pragma unroll
      for (int bi = 0; bi < 2; ++bi) {
        const float px = pv[bi * 5 + 0], py = pv[bi * 5 + 1];
        const float pw = pv[bi * 5 + 2], ph = pv[bi * 5 + 3];
        const float px1 = px / 14.0f - 0.5f * pw;
        const float py1 = py / 14.0f - 0.5f * ph;
        const float px2 = px / 14.0f + 0.5f * pw;
        const float py2 = py / 14.0f + 0.5f * ph;
        const float ltx = fmaxf(px1, tx1), lty = fmaxf(py1, ty1);
        const float rbx = fminf(px2, tx2), rby = fminf(py2, ty2);
        const float wx = fmaxf(rbx - ltx, 0.f), wy = fmaxf(rby - lty, 0.f);
        const float inter = wx * wy;
        const float area_p = (px2 - px1) * (py2 - py1);
        iou[bi] = inter / (area_p + area_t - inter);
      }
      // argmax over 2 boxes, first index on tie (matches jnp.argmax)
      const bool sel = iou[1] > iou[0];
      const float biou = fmaxf(iou[0], iou[1]);
      const float bxv = sel ? pv[5] : pv[0];
      const float byv = sel ? pv[6] : pv[1];
      const float bwv = sel ? pv[7] : pv[2];
      const float bhv = sel ? pv[8] : pv[3];
      const float bcv = sel ? pv[9] : pv[4];
      const float dx = bxv - tx, dy = byv - ty;
      const float dw = sqrtf(bwv) - sqrtf(tw);
      const float dh = sqrtf(bhv) - sqrtf(th);
      s_reg += (dx * dx + dy * dy) + (dw * dw + dh * dh);
      const float dcf = bcv - biou;
      s_con += dcf * dcf;
    } else {
      // conf0 = pred[...,4] = p2[2].x ; conf1 = pred[...,9] = p2[4].y
      const float2 a = p2[2];
      const float2 b = p2[4];
      s_no += a.x * a.x + b.y * b.y;
    }
  }
  block_reduce4(s_reg, s_con, s_no, s_cls, bsums + (size_t)blockIdx.x * 4);
}

__global__ __launch_bounds__(NTHREADS) void yolo_loss_final(
    const float* __restrict__ bsums, int nblocks, float* __restrict__ out,
    float invN) {
  float a0 = 0.f, a1 = 0.f, a2 = 0.f, a3 = 0.f;
  for (int i = threadIdx.x; i < nblocks; i += NTHREADS) {
    const float4 t = reinterpret_cast<const float4*>(bsums)[i];
    a0 += t.x; a1 += t.y; a2 += t.z; a3 += t.w;
  }
  __shared__ float totals[4];
  block_reduce4(a0, a1, a2, a3, totals);
  __syncthreads();
  if (threadIdx.x == 0) {
    const float reg     = (5.0f * totals[0]) * invN;  // L_COORD * sum / N
    const float contain = totals[1] * invN;
    const float no_obj  = (0.5f * totals[2]) * invN;  // L_NOOBJ * sum / N
    const float cls     = totals[3] * invN;
    out[0] = reg + contain + no_obj + cls;
    out[1] = reg;
    out[2] = contain;
    out[3] = no_obj;
    out[4] = cls;
  }
}

extern "C" void kernel_launch(void* const* d_in, const int* in_sizes, int n_in,
                              void* d_out, int out_size, void* d_ws,
                              size_t ws_size, hipStream_t stream) {
  const float* pred         = (const float*)d_in[0];
  const float* tbox         = (const float*)d_in[1];
  const float* tcls         = (const float*)d_in[2];
  const unsigned char* mask = (const unsigned char*)d_in[3];
  float* out   = (float*)d_out;
  float* bsums = (float*)d_ws;  // NBLOCKS*4 floats = 16 KB

  const int ncells = in_sizes[3];          // N*S*S (mask has 1 elem per cell)
  const int ncellsPerImg = 14 * 14;
  const float invN = 1.0f / (float)(ncells / ncellsPerImg);

  yolo_loss_main<<<NBLOCKS, NTHREADS, 0, stream>>>(pred, tbox, tcls, mask,
                                                   bsums, ncells);
  yolo_loss_final<<<1, NTHREADS, 0, stream>>>(bsums, NBLOCKS, out, invN);
}